// LSTMDecoder_61821759258845
// MI455X (gfx1250) — compile-verified
//
#include <hip/hip_runtime.h>
#include <math.h>

// Problem constants (from reference)
#define BB 64
#define TT 25
#define VV 50257
#define EE 512
#define HH 1024
#define GG (4*HH)               // 4096
#define NTILES_V ((VV+15)/16)   // 3142 col tiles for the logits GEMM

typedef __attribute__((ext_vector_type(2))) float v2f;
typedef __attribute__((ext_vector_type(8))) float v8f;

// ---------------------------------------------------------------------------
// V_WMMA_F32_16X16X4_F32 fragment layouts (CDNA5 ISA 7.12.2), wave32:
//   A (16x4):  lanes 0-15 row M=lane, VGPRs = K {0,1}; lanes 16-31 K {2,3}
//   B (4x16):  lanes 0-15 col N=lane, VGPRs = K {0,1}; lanes 16-31 K {2,3}
//   C/D:       VGPR v -> row (v + 8*(lane>=16)), col = lane&15
// Both A and B fragments per lane are the two consecutive K values starting
// at k0 + 2*(lane>=16), so pointers below are pre-offset by 2*half.
// ---------------------------------------------------------------------------

// K-loop with A and B streamed from global (gates kernel: no tail, no guard).
__device__ __forceinline__ v8f kloop_global(const float* __restrict__ arow,
                                            const float* __restrict__ wrow,
                                            int K, v8f acc)
{
    #pragma unroll 8
    for (int k0 = 0; k0 < K; k0 += 4) {
        v2f a = *(const v2f*)(arow + k0);
        v2f b = *(const v2f*)(wrow + k0);
        acc = __builtin_amdgcn_wmma_f32_16x16x4_f32(
                  false, a, false, b, (short)0, acc, false, false);
    }
    return acc;
}

// K-loop with A from LDS (pair-interleaved layout: element (r,k) lives at
// word (k/2)*32 + 2r + (k&1); lane base = &As[32*half + 2r], step 64 words
// per K-step of 4) and B streamed from global.  GUARD=true zeroes B for
// weight rows past V (only the final column tile of the logits GEMM).
template<bool GUARD>
__device__ __forceinline__ v8f kloop_lds(const float* __restrict__ aL,
                                         const float* __restrict__ wrow,
                                         int K, bool valid, v8f acc)
{
    #pragma unroll 8
    for (int k0 = 0; k0 < K; k0 += 4) {
        v2f a = *(const v2f*)(aL + (k0 << 4));   // ds_load_b64, bank-conflict-free
        v2f b = *(const v2f*)(wrow + k0);        // global_load_b64 (W_out stream)
        if (GUARD && !valid) { b.x = 0.0f; b.y = 0.0f; }
        acc = __builtin_amdgcn_wmma_f32_16x16x4_f32(
                  false, a, false, b, (short)0, acc, false, false);
    }
    return acc;
}

// ---------------------------------------------------------------------------
// Step 0: h = conditioning embeddings, c = 0, x = emb_table[SOS=0] broadcast
// ---------------------------------------------------------------------------
__global__ void lstm_init_kernel(const float* __restrict__ emb0,      // [B,H]
                                 const float* __restrict__ emb_table, // [V,E]
                                 float* __restrict__ h,
                                 float* __restrict__ c,
                                 float* __restrict__ x)
{
    int i = blockIdx.x * 256 + threadIdx.x;
    if (i < BB*HH) { h[i] = emb0[i]; c[i] = 0.0f; }
    if (i < BB*EE) { x[i] = emb_table[i % EE]; }   // row 0 repeated per batch
}

// ---------------------------------------------------------------------------
// gates[b,n] = x[b,:]·W_ih[n,:] + h[b,:]·W_hh[n,:] + b_ih[n] + b_hh[n]
// grid = (32, 4), block = 256 (8 waves); wave w -> col tile blockIdx.x*8+w.
// N=4096 is a multiple of 16 -> no guards anywhere.
// ---------------------------------------------------------------------------
__global__ void gates_kernel(const float* __restrict__ x,    // [B,E]
                             const float* __restrict__ h,    // [B,H]
                             const float* __restrict__ Wih,  // [4H,E]
                             const float* __restrict__ Whh,  // [4H,H]
                             const float* __restrict__ bih,
                             const float* __restrict__ bhh,
                             float* __restrict__ gates)      // [B,4H]
{
    const int wave = threadIdx.x >> 5;
    const int lane = threadIdx.x & 31;
    const int half = lane >> 4;
    const int r    = lane & 15;
    const int n0 = (blockIdx.x * 8 + wave) * 16;   // < 4096 always
    const int m0 = blockIdx.y * 16;

    v8f acc = {};
    acc = kloop_global(x + (size_t)(m0 + r)*EE + 2*half,
                       Wih + (size_t)(n0 + r)*EE + 2*half, EE, acc);
    acc = kloop_global(h + (size_t)(m0 + r)*HH + 2*half,
                       Whh + (size_t)(n0 + r)*HH + 2*half, HH, acc);

    const int col = r;
    const float bias = bih[n0 + col] + bhh[n0 + col];
    #pragma unroll
    for (int v = 0; v < 8; ++v)
        gates[(size_t)(m0 + 8*half + v) * GG + n0 + col] = acc[v] + bias;
}

// ---------------------------------------------------------------------------
// LSTM cell elementwise: gate order i,f,g,o (jnp.split order)
// ---------------------------------------------------------------------------
__global__ void cell_kernel(const float* __restrict__ gates,
                            float* __restrict__ h,
                            float* __restrict__ c)
{
    int idx = blockIdx.x * 256 + threadIdx.x;
    if (idx >= BB*HH) return;
    int b = idx / HH, j = idx - b*HH;
    const float* g = gates + (size_t)b * GG;
    float ig = 1.0f / (1.0f + __expf(-g[j]));
    float fg = 1.0f / (1.0f + __expf(-g[HH + j]));
    float gg = tanhf(g[2*HH + j]);
    float og = 1.0f / (1.0f + __expf(-g[3*HH + j]));
    float cn = fg * c[idx] + ig * gg;
    c[idx] = cn;
    h[idx] = og * tanhf(cn);
}

// ---------------------------------------------------------------------------
// logits[b,t,:] = h[b,:]·W_out^T + b_out    (M=64, N=50257, K=1024)
// grid = (393, 4), block = 256 (8 waves).  The block's 16-row A tile (64 KB)
// is staged once into LDS and reused by all 8 column tiles; W_out (206 MB,
// > 192 MB L2) streams from HBM exactly once per step -> bandwidth roofline
// ~8.8 us/step at 23.3 TB/s.
// ---------------------------------------------------------------------------
__global__ void logits_kernel(const float* __restrict__ h,     // [B,H]
                              const float* __restrict__ Wout,  // [V,H]
                              const float* __restrict__ bout,  // [V]
                              float* __restrict__ logits,      // [B,T,V]
                              int t)
{
    // Pair-interleaved A tile: word (k/2)*32 + 2*row + (k&1); 64 KB.
    __shared__ float As[16 * 1024];

    const int m0 = blockIdx.y * 16;

    // Cooperative stage: 8192 v2f elements, conflict-free ds_store_b64.
    for (int i = threadIdx.x; i < 16 * 512; i += 256) {
        int row = i & 15;
        int p   = i >> 4;                               // K-pair index 0..511
        v2f vdat = *(const v2f*)(h + (size_t)(m0 + row)*HH + (p << 1));
        *(v2f*)(&As[p*32 + (row << 1)]) = vdat;
    }
    __syncthreads();

    const int wave = threadIdx.x >> 5;
    const int colTile = blockIdx.x * 8 + wave;
    if (colTile >= NTILES_V) return;                    // uniform per wave
    const int lane = threadIdx.x & 31;
    const int half = lane >> 4;
    const int r    = lane & 15;
    const int n0   = colTile * 16;
    const bool valid = (n0 + r) < VV;

    const float* aL   = &As[half*32 + (r << 1)];
    const float* wrow = Wout + (size_t)(valid ? (n0 + r) : 0)*HH + 2*half;

    // Prefetch the same K-stream of the next column tile (speculative).
    __builtin_prefetch(wrow + (size_t)16 * HH, 0, 0);

    v8f acc = {};
    if (n0 + 16 <= VV) acc = kloop_lds<false>(aL, wrow, HH, true,  acc);
    else               acc = kloop_lds<true >(aL, wrow, HH, valid, acc);

    const int n = n0 + r;
    if (n < VV) {
        const float bias = bout[n];
        #pragma unroll
        for (int v = 0; v < 8; ++v) {
            int row = m0 + 8*half + v;                  // batch index
            logits[((size_t)row * TT + t) * VV + n] = acc[v] + bias;
        }
    }
}

// ---------------------------------------------------------------------------
// Per-row argmax over V (first-max tie-break like jnp.argmax) + gather the
// predicted token's embedding as the next LSTM input.  One block per row.
// ---------------------------------------------------------------------------
__global__ void argmax_gather_kernel(const float* __restrict__ logits, // [B,T,V]
                                     const float* __restrict__ emb,    // [V,E]
                                     float* __restrict__ preds,        // [B,T]
                                     float* __restrict__ x,            // [B,E]
                                     int t)
{
    __shared__ float smax[256];
    __shared__ int   sidx[256];
    __shared__ int   spred;
    const int b = blockIdx.x;
    const float* row = logits + ((size_t)b * TT + t) * VV;

    float best = -INFINITY; int bi = 0;
    for (int v = threadIdx.x; v < VV; v += 256) {
        float val = row[v];
        if (val > best) { best = val; bi = v; }    // strict > keeps lowest idx
    }
    smax[threadIdx.x] = best; sidx[threadIdx.x] = bi;
    __syncthreads();
    for (int s = 128; s > 0; s >>= 1) {
        if (threadIdx.x < s) {
            float ov = smax[threadIdx.x + s]; int oi = sidx[threadIdx.x + s];
            float mv = smax[threadIdx.x];     int mi = sidx[threadIdx.x];
            if (ov > mv || (ov == mv && oi < mi)) {
                smax[threadIdx.x] = ov; sidx[threadIdx.x] = oi;
            }
        }
        __syncthreads();
    }
    if (threadIdx.x == 0) { spred = sidx[0]; preds[b*TT + t] = (float)sidx[0]; }
    __syncthreads();
    const int p = spred;
    for (int e = threadIdx.x; e < EE; e += 256)
        x[(size_t)b*EE + e] = emb[(size_t)p*EE + e];
}

// ---------------------------------------------------------------------------
extern "C" void kernel_launch(void* const* d_in, const int* in_sizes, int n_in,
                              void* d_out, int out_size, void* d_ws, size_t ws_size,
                              hipStream_t stream)
{
    // setup_inputs order: labels, embeddings, emb_table, W_ih, W_hh, b_ih, b_hh, W_out, b_out
    const float* embeddings = (const float*)d_in[1];   // [B,H]
    const float* emb_table  = (const float*)d_in[2];   // [V,E]
    const float* Wih        = (const float*)d_in[3];   // [4H,E]
    const float* Whh        = (const float*)d_in[4];   // [4H,H]
    const float* bih        = (const float*)d_in[5];
    const float* bhh        = (const float*)d_in[6];
    const float* Wout       = (const float*)d_in[7];   // [V,H]
    const float* bout       = (const float*)d_in[8];

    float* preds  = (float*)d_out;        // [B,T] (preds cast to output dtype)
    float* logits = preds + BB*TT;        // [B,T,V]

    // Workspace: h | c | x | gates  (~1.7 MB)
    float* h     = (float*)d_ws;
    float* c     = h + BB*HH;
    float* x     = c + BB*HH;
    float* gates = x + BB*EE;

    lstm_init_kernel<<<(BB*HH + 255)/256, 256, 0, stream>>>(embeddings, emb_table, h, c, x);

    for (int t = 0; t < TT; ++t) {
        gates_kernel<<<dim3(GG/16/8, BB/16), 256, 0, stream>>>(x, h, Wih, Whh, bih, bhh, gates);
        cell_kernel<<<(BB*HH + 255)/256, 256, 0, stream>>>(gates, h, c);
        logits_kernel<<<dim3((NTILES_V + 7)/8, BB/16), 256, 0, stream>>>(h, Wout, bout, logits, t);
        argmax_gather_kernel<<<BB, 256, 0, stream>>>(logits, emb_table, preds, x, t);
    }
}